// KVCache_67224828117321
// MI455X (gfx1250) — compile-verified
//
#include <hip/hip_runtime.h>
#include <stdint.h>

// KV-cache append for MI455X (gfx1250, wave32).
// Pure streaming problem: ~512MB read + ~512MB write => ~44us floor @ 23.3TB/s.
// One fused pass: each thread moves one float4 of K and one of V, selecting its
// source (old cache vs appended values) branchlessly. All bulk traffic is
// non-temporal b128 to stream past the 192MB L2. pos[b] is fetched via the
// scalar path (b is block-uniform) so no per-wave VMEM stall precedes the
// main loads.

typedef float v4f __attribute__((ext_vector_type(4)));

namespace {
constexpr unsigned Bc = 8, Hc = 32, Tc = 2048, Dc = 128, Sc = 512;
constexpr unsigned V4_ROW = Dc / 4;                        // 32 float4 per row
constexpr unsigned ROWS   = Bc * Hc * Tc;                  // 524288 rows per cache
constexpr unsigned NV4    = ROWS * V4_ROW;                 // 16,777,216 float4 per cache
constexpr unsigned long long CACHE_ELTS =
    (unsigned long long)ROWS * Dc;                         // 67,108,864 floats per cache
}

__global__ __launch_bounds__(256) void kv_append_kernel(
    const v4f* __restrict__ k_cache, const v4f* __restrict__ v_cache,
    const int* __restrict__ pos,
    const v4f* __restrict__ k_val,   const v4f* __restrict__ v_val,
    v4f* __restrict__ out_k, v4f* __restrict__ out_v, int* __restrict__ out_pos)
{
    unsigned idx   = blockIdx.x * 256u + threadIdx.x;   // < 2^24, one float4 of K and V
    unsigned lane4 = idx & (V4_ROW - 1);                // float4 index within D
    unsigned row   = idx >> 5;                          // b*H*T + h*T + t
    unsigned t     = row & (Tc - 1);
    unsigned bh    = row >> 11;                         // b*H + h

    // b is uniform across the block: 2^21 float4s per batch element, 256 per
    // block => deriving it from blockIdx.x keeps the pos[] fetch scalar
    // (s_load_b32 via KMcnt) instead of a per-lane VMEM load + loadcnt stall.
    unsigned b = blockIdx.x >> 13;                      // == idx >> 21

    int p = pos[b];
    unsigned rel   = t - (unsigned)p;                   // unsigned wrap => single compare
    bool fresh     = rel < Sc;                          // row is inside the appended slice

    // val layout: [B,H,S,D] -> float4 index = (bh*S + rel)*32 + lane4
    unsigned src = fresh ? (((bh * Sc + rel) << 5) | lane4) : idx;
    const v4f* __restrict__ ks = fresh ? k_val : k_cache;
    const v4f* __restrict__ vs = fresh ? v_val : v_cache;

    // Streaming: read-once / write-once, bypass temporal caching.
    v4f kd = __builtin_nontemporal_load(&ks[src]);
    v4f vd = __builtin_nontemporal_load(&vs[src]);
    __builtin_nontemporal_store(kd, &out_k[idx]);
    __builtin_nontemporal_store(vd, &out_v[idx]);

    if (idx < Bc) out_pos[idx] = pos[idx] + (int)Sc;    // new_pos = pos + S
}

// ---------------------------------------------------------------------------
// CDNA5 async-LDS bulk-copy path (assembles cleanly on gfx1250; NOT launched).
// GLOBAL_LOAD_ASYNC_TO_LDS_B128 -> s_wait_asynccnt -> GLOBAL_STORE_ASYNC_FROM_LDS_B128
// For a pure global->global stream this adds an LDS hop with the same
// issue/wait structure as the direct NT b128 path, so the direct path is used
// for the real work; this kernel documents/exercises the ASYNCcnt machinery.
// ---------------------------------------------------------------------------
__global__ void kv_async_probe(const float* __restrict__ gsrc,
                               float* __restrict__ gdst)
{
    __shared__ v4f stage[256];
    unsigned tid = threadIdx.x;
    // Workgroup-relative LDS byte address for this lane's 16B slot.
    unsigned lds_addr = (unsigned)(uintptr_t)
        (__attribute__((address_space(3))) v4f*)&stage[tid];

    const float* s = gsrc + (blockIdx.x * 256u + tid) * 4u;
    float*       d = gdst + (blockIdx.x * 256u + tid) * 4u;

    // memory -> LDS (ASYNCcnt)
    asm volatile("global_load_async_to_lds_b128 %0, %1, off"
                 :: "v"(lds_addr), "v"(s) : "memory");
    // drain our async loads before the dependent LDS read
    asm volatile("s_wait_asynccnt 0" ::: "memory");
    // LDS -> memory (ASYNCcnt)
    asm volatile("global_store_async_from_lds_b128 %0, %1, off"
                 :: "v"(d), "v"(lds_addr) : "memory");
    asm volatile("s_wait_asynccnt 0" ::: "memory");
}

extern "C" void kernel_launch(void* const* d_in, const int* in_sizes, int n_in,
                              void* d_out, int out_size, void* d_ws, size_t ws_size,
                              hipStream_t stream) {
    // setup_inputs order: k_cache, v_cache, current_pos, k_val, v_val
    const v4f* k_cache = (const v4f*)d_in[0];
    const v4f* v_cache = (const v4f*)d_in[1];
    const int* pos     = (const int*)d_in[2];
    const v4f* k_val   = (const v4f*)d_in[3];
    const v4f* v_val   = (const v4f*)d_in[4];

    float* out   = (float*)d_out;
    v4f* out_k   = (v4f*)out;                            // [B,H,T,D]
    v4f* out_v   = (v4f*)(out + CACHE_ELTS);             // [B,H,T,D]
    int* out_pos = (int*)(out + 2ull * CACHE_ELTS);      // [B] int32 bit pattern

    kv_append_kernel<<<NV4 / 256, 256, 0, stream>>>(
        k_cache, v_cache, pos, k_val, v_val, out_k, out_v, out_pos);
}